// Regression_X1_Up_66013647339744
// MI455X (gfx1250) — compile-verified
//
#include <hip/hip_runtime.h>
#include <hip/hip_bf16.h>

typedef float v2f __attribute__((ext_vector_type(2)));
typedef float v8f __attribute__((ext_vector_type(8)));

#define N_NODES 50000
#define N_EDGES 800000
#define H_DIM   64
#define NUM_RELS 4

// ---------------------------------------------------------------------------
// Basis composition, faithful to the torch raw-buffer reinterpret:
// weight (B=2, I, O) flat; wt = reinterpret as (O, B, I);
// out[(o,r,i)] = sum_b comp[r][b] * wt[o][b][i], written flat in (o,r,i) order.
// Consumers then read the SAME buffer with (r,o,i) indexing (raw reshape).
// ---------------------------------------------------------------------------
__global__ void compose_kernel(const float* __restrict__ weight,
                               const float* __restrict__ comp,   // (4, 2)
                               float* __restrict__ out,
                               int O, int I) {
    int total = O * NUM_RELS * I;
    for (int idx = blockIdx.x * blockDim.x + threadIdx.x; idx < total;
         idx += gridDim.x * blockDim.x) {
        int o   = idx / (NUM_RELS * I);
        int rem = idx - o * NUM_RELS * I;
        int r   = rem / I;
        int i   = rem - r * I;
        const float* wb = weight + (size_t)o * 2 * I + i;  // wt[o][b][i]
        out[idx] = comp[r * 2 + 0] * wb[0] + comp[r * 2 + 1] * wb[I];
    }
}

__global__ void zero_kernel(float* __restrict__ p, size_t n) {
    size_t i = (size_t)blockIdx.x * blockDim.x + threadIdx.x;
    if (i < n) p[i] = 0.0f;
}

// ---------------------------------------------------------------------------
// proj[r][m][n] = sum_k relu?(H[m][k]) * W[r][n][k]   using V_WMMA_F32_16X16X4_F32
// H: (M, K) row-major. W buffer read as (R, 64, K). proj: (R, M, 64).
// Block = 128 threads (4 waves); blockIdx.x = M-tile (16 rows), blockIdx.y = r;
// wave w computes N-tile n0 = 16*w.
// ---------------------------------------------------------------------------
template <int K, bool RELU>
__global__ void __launch_bounds__(128)
rgcn_proj_wmma(const float* __restrict__ H,
               const float* __restrict__ W,
               float* __restrict__ proj,
               int M) {
    const int lane  = threadIdx.x & 31;
    const int wave  = threadIdx.x >> 5;
    const int m0    = blockIdx.x * 16;
    const int r     = blockIdx.y;
    const int n0    = wave * 16;
    const int l15   = lane & 15;
    const int khalf = (lane >> 4) * 2;   // lanes 0-15: K+0,1 ; lanes 16-31: K+2,3

    const float* Arow = H + (size_t)(m0 + l15) * K + khalf;
    const float* Brow = W + (size_t)r * H_DIM * K + (size_t)(n0 + l15) * K + khalf;

    v8f acc = {};
#pragma unroll
    for (int kk = 0; kk < K; kk += 4) {
        v2f a = *(const v2f*)(Arow + kk);
        v2f b = *(const v2f*)(Brow + kk);
        if (RELU) { a.x = fmaxf(a.x, 0.0f); a.y = fmaxf(a.y, 0.0f); }
        acc = __builtin_amdgcn_wmma_f32_16x16x4_f32(false, a, false, b,
                                                    (short)0, acc, false, false);
    }

    // C/D layout: VGPR v -> row m0 + v + 8*(lane>>4), col n0 + (lane&15)
    float* out = proj + ((size_t)r * M + m0) * H_DIM + n0;
    const int rowoff = (lane >> 4) * 8;
#pragma unroll
    for (int v = 0; v < 8; ++v) {
        out[(size_t)(v + rowoff) * H_DIM + l15] = acc[v];
    }
}

// ---------------------------------------------------------------------------
// Edge scatter: agg[dst[e]][f] += proj[rel[e]][src[e]][f], 64 threads per edge.
// ---------------------------------------------------------------------------
__global__ void scatter_kernel(const float* __restrict__ proj,
                               const int* __restrict__ src,
                               const int* __restrict__ dst,
                               const int* __restrict__ rel,
                               float* __restrict__ agg,
                               int E, int M) {
    long long t = (long long)blockIdx.x * blockDim.x + threadIdx.x;
    if (t >= (long long)E * H_DIM) return;
    int e = (int)(t >> 6);
    int f = (int)(t & 63);
    int s = src[e];
    int d = dst[e];
    int r = rel[e];
    float val = proj[((size_t)r * M + s) * H_DIM + f];
    unsafeAtomicAdd(&agg[(size_t)d * H_DIM + f], val);
}

// ---------------------------------------------------------------------------
// Output layer projection: s[r][n] = sum_i relu(agg2[n][i]) * w3[r*64 + i]
// ---------------------------------------------------------------------------
__global__ void outproj_kernel(const float* __restrict__ agg2,
                               const float* __restrict__ w3,
                               float* __restrict__ s, int M) {
    int n = blockIdx.x * blockDim.x + threadIdx.x;
    if (n >= M) return;
    float a0 = 0.f, a1 = 0.f, a2 = 0.f, a3 = 0.f;
    const float* row = agg2 + (size_t)n * H_DIM;
#pragma unroll 4
    for (int i = 0; i < H_DIM; ++i) {
        float h = fmaxf(row[i], 0.0f);
        a0 += w3[i]       * h;
        a1 += w3[64 + i]  * h;
        a2 += w3[128 + i] * h;
        a3 += w3[192 + i] * h;
    }
    s[n]         = a0;
    s[M + n]     = a1;
    s[2 * M + n] = a2;
    s[3 * M + n] = a3;
}

// out[e] = s[rel[e]][src[e]]
__global__ void gather_kernel(const float* __restrict__ s,
                              const int* __restrict__ src,
                              const int* __restrict__ rel,
                              float* __restrict__ out, int E, int M) {
    int e = blockIdx.x * blockDim.x + threadIdx.x;
    if (e >= E) return;
    out[e] = s[(size_t)rel[e] * M + src[e]];
}

extern "C" void kernel_launch(void* const* d_in, const int* in_sizes, int n_in,
                              void* d_out, int out_size, void* d_ws, size_t ws_size,
                              hipStream_t stream) {
    const float* feature    = (const float*)d_in[0];
    const int*   src        = (const int*)d_in[1];
    const int*   dst        = (const int*)d_in[2];
    const int*   rel        = (const int*)d_in[3];
    const float* weight_in  = (const float*)d_in[4];
    const float* w_comp_in  = (const float*)d_in[5];
    const float* weight_h0  = (const float*)d_in[6];
    const float* w_comp_h0  = (const float*)d_in[7];
    const float* weight_out = (const float*)d_in[8];
    const float* w_comp_out = (const float*)d_in[9];
    float*       out        = (float*)d_out;

    const int M = N_NODES, E = N_EDGES;

    // Workspace layout (floats). Total ~19.4M floats ~= 78 MB.
    float* ws   = (float*)d_ws;
    float* w1   = ws;                                 // 64*4*16   = 4096
    float* w2   = w1 + 4096;                          // 64*4*64   = 16384
    float* w3   = w2 + 16384;                         // 1*4*64    = 256
    float* proj = w3 + 256;                           // 4*M*64    = 12.8M (reused by both layers)
    float* agg1 = proj + (size_t)NUM_RELS * M * H_DIM; // M*64
    float* agg2 = agg1 + (size_t)M * H_DIM;            // M*64 (contiguous with agg1)
    float* sbuf = agg2 + (size_t)M * H_DIM;            // 4*M

    // 1) Compose relation weights from bases (tiny).
    compose_kernel<<<16, 256, 0, stream>>>(weight_in,  w_comp_in,  w1, H_DIM, 16);
    compose_kernel<<<64, 256, 0, stream>>>(weight_h0,  w_comp_h0,  w2, H_DIM, H_DIM);
    compose_kernel<<<1,  256, 0, stream>>>(weight_out, w_comp_out, w3, 1,     H_DIM);

    // 2) Zero both accumulators (contiguous region) — required every launch.
    size_t nz = (size_t)2 * M * H_DIM;
    zero_kernel<<<(int)((nz + 255) / 256), 256, 0, stream>>>(agg1, nz);

    dim3 gemm_grid(M / 16, NUM_RELS);
    int  scat_blocks = (int)(((long long)E * H_DIM + 255) / 256);

    // 3) Layer 1: proj1[r] = feature @ W_in[r]^T  (K=16), scatter to agg1.
    rgcn_proj_wmma<16, false><<<gemm_grid, 128, 0, stream>>>(feature, w1, proj, M);
    scatter_kernel<<<scat_blocks, 256, 0, stream>>>(proj, src, dst, rel, agg1, E, M);

    // 4) Layer 2: proj2[r] = relu(agg1) @ W_h0[r]^T  (K=64), scatter to agg2.
    rgcn_proj_wmma<64, true><<<gemm_grid, 128, 0, stream>>>(agg1, w2, proj, M);
    scatter_kernel<<<scat_blocks, 256, 0, stream>>>(proj, src, dst, rel, agg2, E, M);

    // 5) Output layer: per-node scalar per relation, then per-edge gather.
    outproj_kernel<<<(M + 255) / 256, 256, 0, stream>>>(agg2, w3, sbuf, M);
    gather_kernel<<<(E + 255) / 256, 256, 0, stream>>>(sbuf, src, rel, out, E, M);
}